// GAT_29506425323821
// MI455X (gfx1250) — compile-verified
//
#include <hip/hip_runtime.h>
#include <hip/hip_bf16.h>
#include <math.h>

#define NNODES 50000
#define EDGES  800000
#define INFEAT 128
#define HIDTOT 256   // 4 heads * 64
#define NEG_SLOPE 0.2f
#define ORD_NEG_INF 0x007FFFFFu   // ford(-inf)

typedef __attribute__((ext_vector_type(2))) float v2f;
typedef __attribute__((ext_vector_type(8))) float v8f;

// ---- order-preserving float <-> uint for atomicMax on signed floats ----
__device__ __forceinline__ unsigned ford(float f) {
    unsigned u = __float_as_uint(f);
    return (u & 0x80000000u) ? ~u : (u | 0x80000000u);
}
__device__ __forceinline__ float fdec(unsigned u) {
    return __uint_as_float((u & 0x80000000u) ? (u & 0x7FFFFFFFu) : ~u);
}
__device__ __forceinline__ float lrelu(float x) {
    return x >= 0.f ? x : NEG_SLOPE * x;
}

// ---------------------------------------------------------------------------
// K0: reset accumulators (runs first every call -> deterministic replays)
// ---------------------------------------------------------------------------
__global__ void init_kernel(float* out1, float* denom1, unsigned* m1,
                            float* denom2, unsigned* m2, float* dout) {
    size_t i = (size_t)blockIdx.x * blockDim.x + threadIdx.x;
    if (i < (size_t)NNODES * HIDTOT) out1[i] = 0.f;
    if (i < (size_t)NNODES * 4) { denom1[i] = 0.f; m1[i] = ORD_NEG_INF; }
    if (i < (size_t)NNODES)     { denom2[i] = 0.f; m2[i] = ORD_NEG_INF; }
    if (i < (size_t)NNODES * 2) dout[i] = 0.f;
}

// ---------------------------------------------------------------------------
// K1: fused dual GEMM: hs = feat@Ws + bs ; hd = feat@Wd + bd
//     fp32 WMMA 16x16x4. Each wave: one 16-row tile x 64 output cols,
//     A-fragment reused across 4 column tiles and across both weight matrices.
// ---------------------------------------------------------------------------
__global__ __launch_bounds__(256) void gemm1_kernel(
    const float* __restrict__ feat,
    const float* __restrict__ Ws, const float* __restrict__ bs,
    const float* __restrict__ Wd, const float* __restrict__ bd,
    float* __restrict__ hs, float* __restrict__ hd)
{
    const int wave = threadIdx.x >> 5;
    const int lane = threadIdx.x & 31;
    const int half = lane >> 4;     // K-halves for A, row-halves for C
    const int row  = lane & 15;

    int tile = blockIdx.x * 8 + wave;           // 3125 M-tiles * 4 col-groups
    if (tile >= 3125 * 4) return;
    const int mt  = tile >> 2;
    const int nt4 = tile & 3;
    const int m0  = mt * 16;
    const int c0  = nt4 * 64;

    v8f zero = {0.f,0.f,0.f,0.f,0.f,0.f,0.f,0.f};
    v8f accS[4], accD[4];
    #pragma unroll
    for (int j = 0; j < 4; ++j) { accS[j] = zero; accD[j] = zero; }

    const float* arow = feat + (size_t)(m0 + row) * INFEAT;
    for (int k0 = 0; k0 < INFEAT; k0 += 4) {
        const int ka = k0 + 2 * half;           // ISA 16x4 f32 A layout
        v2f a = *(const v2f*)(arow + ka);
        #pragma unroll
        for (int j = 0; j < 4; ++j) {
            const int col = c0 + j * 16 + row;  // ISA 4x16 B layout
            v2f bS, bD;
            bS.x = Ws[(size_t)ka * HIDTOT + col];
            bS.y = Ws[(size_t)(ka + 1) * HIDTOT + col];
            bD.x = Wd[(size_t)ka * HIDTOT + col];
            bD.y = Wd[(size_t)(ka + 1) * HIDTOT + col];
            accS[j] = __builtin_amdgcn_wmma_f32_16x16x4_f32(
                false, a, false, bS, (short)0, accS[j], false, false);
            accD[j] = __builtin_amdgcn_wmma_f32_16x16x4_f32(
                false, a, false, bD, (short)0, accD[j], false, false);
        }
    }
    #pragma unroll
    for (int j = 0; j < 4; ++j) {
        const int col = c0 + j * 16 + row;
        const float biasS = bs[col], biasD = bd[col];
        #pragma unroll
        for (int i = 0; i < 8; ++i) {           // C rows: i (lanes 0-15), i+8
            const int r = m0 + i + 8 * half;
            hs[(size_t)r * HIDTOT + col] = accS[j][i] + biasS;
            hd[(size_t)r * HIDTOT + col] = accD[j][i] + biasD;
        }
    }
}

// ---------------------------------------------------------------------------
// K2: per-edge scores + segment max (one wave per edge, 8 dims per lane)
// ---------------------------------------------------------------------------
__global__ __launch_bounds__(256) void edge_score1_kernel(
    const float* __restrict__ hs, const float* __restrict__ hd,
    const int* __restrict__ src, const int* __restrict__ dst,
    const float* __restrict__ attn, float* __restrict__ score,
    unsigned* __restrict__ mbits)
{
    const int e = blockIdx.x * 8 + (threadIdx.x >> 5);
    if (e >= EDGES) return;
    const int lane = threadIdx.x & 31;
    const int s = src[e], d = dst[e];
    const int base = lane * 8;

    const float4* ph = (const float4*)(hs + (size_t)s * HIDTOT + base);
    const float4* pd = (const float4*)(hd + (size_t)d * HIDTOT + base);
    const float4* pa = (const float4*)(attn + base);

    float part = 0.f;
    #pragma unroll
    for (int q = 0; q < 2; ++q) {
        float4 a = ph[q], b = pd[q], w = pa[q];
        part += lrelu(a.x + b.x) * w.x + lrelu(a.y + b.y) * w.y +
                lrelu(a.z + b.z) * w.z + lrelu(a.w + b.w) * w.w;
    }
    // reduce each 8-lane group (= one 64-dim head)
    part += __shfl_xor(part, 1, 32);
    part += __shfl_xor(part, 2, 32);
    part += __shfl_xor(part, 4, 32);
    if ((lane & 7) == 0) {
        const int h = lane >> 3;
        score[(size_t)e * 4 + h] = part;
        atomicMax(mbits + (size_t)d * 4 + h, ford(part));
    }
}

// ---------------------------------------------------------------------------
// K3: exp(score - m) + segment sum; score buffer is overwritten with ex
// ---------------------------------------------------------------------------
__global__ void edge_exp1_kernel(const int* __restrict__ dst,
                                 float* __restrict__ score,
                                 const unsigned* __restrict__ mbits,
                                 float* __restrict__ denom)
{
    const int t = blockIdx.x * blockDim.x + threadIdx.x;
    if (t >= EDGES * 4) return;
    const int e = t >> 2, h = t & 3;
    const int d = dst[e];
    float m = fdec(mbits[(size_t)d * 4 + h]);
    if (!isfinite(m)) m = 0.f;
    const float ex = expf(score[t] - m);
    score[t] = ex;
    atomicAdd(denom + (size_t)d * 4 + h, ex);
}

// ---------------------------------------------------------------------------
// K4: alpha-weighted scatter: out1[dst] += alpha * hs[src]
// ---------------------------------------------------------------------------
__global__ __launch_bounds__(256) void edge_aggr1_kernel(
    const float* __restrict__ hs, const int* __restrict__ src,
    const int* __restrict__ dst, const float* __restrict__ ex,
    const float* __restrict__ denom, float* __restrict__ out)
{
    const int e = blockIdx.x * 8 + (threadIdx.x >> 5);
    if (e >= EDGES) return;
    const int lane = threadIdx.x & 31;
    const int s = src[e], d = dst[e];
    const int h = lane >> 3;
    float den = denom[(size_t)d * 4 + h];
    den = den > 0.f ? den : 1.f;
    const float alpha = ex[(size_t)e * 4 + h] / den;
    const int base = lane * 8;
    const float4* ph = (const float4*)(hs + (size_t)s * HIDTOT + base);
    float* po = out + (size_t)d * HIDTOT + base;
    #pragma unroll
    for (int q = 0; q < 2; ++q) {
        float4 a = ph[q];
        atomicAdd(po + q * 4 + 0, alpha * a.x);
        atomicAdd(po + q * 4 + 1, alpha * a.y);
        atomicAdd(po + q * 4 + 2, alpha * a.z);
        atomicAdd(po + q * 4 + 3, alpha * a.w);
    }
}

// ---------------------------------------------------------------------------
// K5: ELU in place
// ---------------------------------------------------------------------------
__global__ void elu_kernel(float* __restrict__ x, int n) {
    const int i = blockIdx.x * blockDim.x + threadIdx.x;
    if (i < n) {
        const float v = x[i];
        x[i] = v > 0.f ? v : (expf(v) - 1.f);
    }
}

// ---------------------------------------------------------------------------
// K6: layer-2 projections (wave per node, shuffle reduction; out dim = 2)
// ---------------------------------------------------------------------------
__global__ __launch_bounds__(256) void proj2_kernel(
    const float* __restrict__ h,
    const float* __restrict__ W2s, const float* __restrict__ b2s,
    const float* __restrict__ W2d, const float* __restrict__ b2d,
    float* __restrict__ hs2, float* __restrict__ hd2)
{
    const int n = blockIdx.x * 8 + (threadIdx.x >> 5);
    if (n >= NNODES) return;
    const int lane = threadIdx.x & 31;
    const int base = lane * 8;
    const float* hr = h + (size_t)n * HIDTOT + base;
    float s0 = 0.f, s1 = 0.f, d0 = 0.f, d1 = 0.f;
    #pragma unroll
    for (int k = 0; k < 8; ++k) {
        const float x = hr[k];
        const float2 ws = *(const float2*)(W2s + (size_t)(base + k) * 2);
        const float2 wd = *(const float2*)(W2d + (size_t)(base + k) * 2);
        s0 += x * ws.x; s1 += x * ws.y;
        d0 += x * wd.x; d1 += x * wd.y;
    }
    #pragma unroll
    for (int o = 1; o < 32; o <<= 1) {
        s0 += __shfl_xor(s0, o, 32); s1 += __shfl_xor(s1, o, 32);
        d0 += __shfl_xor(d0, o, 32); d1 += __shfl_xor(d1, o, 32);
    }
    if (lane == 0) {
        hs2[(size_t)n * 2 + 0] = s0 + b2s[0];
        hs2[(size_t)n * 2 + 1] = s1 + b2s[1];
        hd2[(size_t)n * 2 + 0] = d0 + b2d[0];
        hd2[(size_t)n * 2 + 1] = d1 + b2d[1];
    }
}

// ---------------------------------------------------------------------------
// K7/K8/K9: layer-2 edge phase (D=2, single head) -> accumulate into d_out
// ---------------------------------------------------------------------------
__global__ void edge_score2_kernel(const float* __restrict__ hs2,
                                   const float* __restrict__ hd2,
                                   const int* __restrict__ src,
                                   const int* __restrict__ dst,
                                   const float* __restrict__ attn2,
                                   float* __restrict__ score2,
                                   unsigned* __restrict__ m2)
{
    const int e = blockIdx.x * blockDim.x + threadIdx.x;
    if (e >= EDGES) return;
    const int s = src[e], d = dst[e];
    const float2 a = *(const float2*)(hs2 + (size_t)s * 2);
    const float2 b = *(const float2*)(hd2 + (size_t)d * 2);
    const float sc = lrelu(a.x + b.x) * attn2[0] + lrelu(a.y + b.y) * attn2[1];
    score2[e] = sc;
    atomicMax(m2 + d, ford(sc));
}

__global__ void edge_exp2_kernel(const int* __restrict__ dst,
                                 float* __restrict__ score2,
                                 const unsigned* __restrict__ m2,
                                 float* __restrict__ denom2)
{
    const int e = blockIdx.x * blockDim.x + threadIdx.x;
    if (e >= EDGES) return;
    const int d = dst[e];
    float m = fdec(m2[d]);
    if (!isfinite(m)) m = 0.f;
    const float ex = expf(score2[e] - m);
    score2[e] = ex;
    atomicAdd(denom2 + d, ex);
}

__global__ void edge_aggr2_kernel(const float* __restrict__ hs2,
                                  const int* __restrict__ src,
                                  const int* __restrict__ dst,
                                  const float* __restrict__ ex,
                                  const float* __restrict__ denom2,
                                  float* __restrict__ out)
{
    const int e = blockIdx.x * blockDim.x + threadIdx.x;
    if (e >= EDGES) return;
    const int s = src[e], d = dst[e];
    float den = denom2[d];
    den = den > 0.f ? den : 1.f;
    const float alpha = ex[e] / den;
    const float2 a = *(const float2*)(hs2 + (size_t)s * 2);
    atomicAdd(out + (size_t)d * 2 + 0, alpha * a.x);
    atomicAdd(out + (size_t)d * 2 + 1, alpha * a.y);
}

// ---------------------------------------------------------------------------
extern "C" void kernel_launch(void* const* d_in, const int* in_sizes, int n_in,
                              void* d_out, int out_size, void* d_ws, size_t ws_size,
                              hipStream_t stream)
{
    const float* feat  = (const float*)d_in[0];
    const int*   src   = (const int*)d_in[1];
    const int*   dst   = (const int*)d_in[2];
    const float* W1s   = (const float*)d_in[3];
    const float* b1s   = (const float*)d_in[4];
    const float* W1d   = (const float*)d_in[5];
    const float* b1d   = (const float*)d_in[6];
    const float* attn1 = (const float*)d_in[7];
    const float* W2s   = (const float*)d_in[8];
    const float* b2s   = (const float*)d_in[9];
    const float* W2d   = (const float*)d_in[10];
    const float* b2d   = (const float*)d_in[11];
    const float* attn2 = (const float*)d_in[12];
    float* out = (float*)d_out;

    // workspace carve-up (all float-sized slots)
    float* ws = (float*)d_ws;
    size_t off = 0;
    float*    hs1    = ws + off; off += (size_t)NNODES * HIDTOT;
    float*    hd1    = ws + off; off += (size_t)NNODES * HIDTOT;
    float*    out1   = ws + off; off += (size_t)NNODES * HIDTOT;
    float*    score1 = ws + off; off += (size_t)EDGES * 4;
    float*    denom1 = ws + off; off += (size_t)NNODES * 4;
    unsigned* m1     = (unsigned*)(ws + off); off += (size_t)NNODES * 4;
    float*    hs2    = ws + off; off += (size_t)NNODES * 2;
    float*    hd2    = ws + off; off += (size_t)NNODES * 2;
    float*    score2 = ws + off; off += (size_t)EDGES;
    float*    denom2 = ws + off; off += (size_t)NNODES;
    unsigned* m2     = (unsigned*)(ws + off); off += (size_t)NNODES;

    const int B = 256;

    // K0: reset accumulators / maxes / output
    {
        size_t n = (size_t)NNODES * HIDTOT;
        init_kernel<<<(unsigned)((n + B - 1) / B), B, 0, stream>>>(
            out1, denom1, m1, denom2, m2, out);
    }
    // K1: dual WMMA GEMM
    {
        const int tiles = 3125 * 4;
        gemm1_kernel<<<(tiles + 7) / 8, B, 0, stream>>>(
            feat, W1s, b1s, W1d, b1d, hs1, hd1);
    }
    // K2..K4: layer-1 edge phase
    edge_score1_kernel<<<(EDGES + 7) / 8, B, 0, stream>>>(
        hs1, hd1, src, dst, attn1, score1, m1);
    edge_exp1_kernel<<<(EDGES * 4 + B - 1) / B, B, 0, stream>>>(
        dst, score1, m1, denom1);
    edge_aggr1_kernel<<<(EDGES + 7) / 8, B, 0, stream>>>(
        hs1, src, dst, score1, denom1, out1);
    // K5: ELU
    elu_kernel<<<((NNODES * HIDTOT) + B - 1) / B, B, 0, stream>>>(
        out1, NNODES * HIDTOT);
    // K6: layer-2 projections
    proj2_kernel<<<(NNODES + 7) / 8, B, 0, stream>>>(
        out1, W2s, b2s, W2d, b2d, hs2, hd2);
    // K7..K9: layer-2 edge phase, accumulate into d_out
    edge_score2_kernel<<<(EDGES + B - 1) / B, B, 0, stream>>>(
        hs2, hd2, src, dst, attn2, score2, m2);
    edge_exp2_kernel<<<(EDGES + B - 1) / B, B, 0, stream>>>(
        dst, score2, m2, denom2);
    edge_aggr2_kernel<<<(EDGES + B - 1) / B, B, 0, stream>>>(
        hs2, src, dst, score2, denom2, out);

    (void)in_sizes; (void)n_in; (void)out_size; (void)ws_size;
}